// SpikingNeuralNetwork_82617990906022
// MI455X (gfx1250) — compile-verified
//
#include <hip/hip_runtime.h>

typedef __attribute__((ext_vector_type(2))) float v2f;
typedef __attribute__((ext_vector_type(8))) float v8f;

#define BATCH  32
#define SEQ    4096
#define EMBED  512
#define HIDDEN 256
#define OUTD   128

// ---------------------------------------------------------------------------
// Phase 1: batch-mean over b. Reads 256 MiB -> the only HBM-bound phase.
// Coalesced: consecutive lanes read consecutive (s,d) elements; the 32 b-slices
// are independent streams the compiler can keep in flight.
// ---------------------------------------------------------------------------
__global__ __launch_bounds__(256) void k_batch_mean(const float* __restrict__ x,
                                                    float* __restrict__ xm) {
  size_t i = (size_t)blockIdx.x * 256u + threadIdx.x;   // over SEQ*EMBED = 2M
  float s = 0.f;
  #pragma unroll
  for (int b = 0; b < BATCH; ++b) {
    s += x[(size_t)b * ((size_t)SEQ * EMBED) + i];
  }
  xm[i] = s * (1.0f / BATCH);
}

// ---------------------------------------------------------------------------
// Phase 2: currents[4096,256] = xm[4096,512] @ W1[512,256] via
// V_WMMA_F32_16X16X4_F32 (f32 in, f32 accum -> no precision loss vs reference).
// One wave per 16x16 output tile; 512/4 = 128 WMMAs per wave.
// A layout (ISA 7.12.2, 32-bit A 16x4): lane m = l&15, half = l>>4,
//   VGPR j holds K = 2*half + j  -> contiguous float2 per lane.
// B layout (row-striped across lanes per VGPR): VGPR v, half h -> K = v + 2h,
//   N = l&15.
// C/D: VGPR v, half h -> M = v + 8h, N = l&15.
// ---------------------------------------------------------------------------
__global__ __launch_bounds__(32) void k_gemm_wmma(const float* __restrict__ xm,
                                                  const float* __restrict__ w1,
                                                  float* __restrict__ cur) {
  const int s0   = blockIdx.x * 16;
  const int h0   = blockIdx.y * 16;
  const int lane = threadIdx.x;
  const int m    = lane & 15;
  const int half = lane >> 4;

  v8f acc = {};
  const float* __restrict__ arow = xm + (size_t)(s0 + m) * EMBED + 2 * half;
  const float* __restrict__ bcol = w1 + h0 + m;

  for (int k = 0; k < EMBED; k += 4) {
    v2f a;
    a.x = arow[k];        // K = k + 2*half + 0
    a.y = arow[k + 1];    // K = k + 2*half + 1
    const int kb = k + 2 * half;
    v2f b;
    b.x = bcol[(size_t)(kb + 0) * HIDDEN];   // K = k + 2h + 0, N = m
    b.y = bcol[(size_t)(kb + 1) * HIDDEN];   // K = k + 2h + 1, N = m
    acc = __builtin_amdgcn_wmma_f32_16x16x4_f32(
        /*neg_a=*/false, a, /*neg_b=*/false, b,
        /*c_mod=*/(short)0, acc, /*reuse_a=*/false, /*reuse_b=*/false);
  }

  #pragma unroll
  for (int v = 0; v < 8; ++v) {
    cur[(size_t)(s0 + v + 8 * half) * HIDDEN + (h0 + m)] = acc[v];
  }
}

// ---------------------------------------------------------------------------
// Phase 3: the spiking scan. Each hidden unit h is an independent chain over
// s = 0..4095; only the spike COUNT matters downstream (spikes are {0,1} and
// the output is a mean over s of spikes @ W2). 256 threads = 8 waves on one
// WGP -> 2 waves/SIMD interleave the per-step dependency chain.
// Per-step math is the reference's exactly, written in select form:
//   active mul  -> cndmask ; mem*(1-spike) -> cndmask ; spikes*2 -> cndmask.
// ---------------------------------------------------------------------------
__global__ __launch_bounds__(256) void k_scan(const float* __restrict__ cur,
                                              float* __restrict__ counts) {
  const int h = threadIdx.x;            // 0..255, coalesced reads per step
  float mem = 0.f, refr = 0.f, cnt = 0.f;
  #pragma unroll 4
  for (int s = 0; s < SEQ; ++s) {
    const float ic  = cur[(size_t)s * HIDDEN + h];
    const float icm = (refr <= 0.f) ? ic : 0.f;   // ic * active
    mem = 0.95f * mem + icm;                      // DECAY
    const bool spike = (mem >= 1.0f);             // THRESHOLD
    cnt  += spike ? 1.f : 0.f;
    refr  = fmaxf(refr - 1.f, 0.f) + (spike ? 2.0f : 0.f);  // REFRACTORY
    mem   = spike ? 0.f : mem;                    // mem * (1 - spikes)
  }
  counts[h] = cnt;
}

// ---------------------------------------------------------------------------
// Phase 4: out_mean[o] = (counts/SEQ) @ W2, broadcast across the batch.
// ---------------------------------------------------------------------------
__global__ __launch_bounds__(128) void k_out(const float* __restrict__ counts,
                                             const float* __restrict__ w2,
                                             float* __restrict__ out) {
  const int o = threadIdx.x;            // 0..127
  float acc = 0.f;
  for (int hh = 0; hh < HIDDEN; ++hh)
    acc += counts[hh] * w2[(size_t)hh * OUTD + o];
  const float val = acc * (1.0f / SEQ);
  #pragma unroll
  for (int b = 0; b < BATCH; ++b)
    out[(size_t)b * OUTD + o] = val;
}

extern "C" void kernel_launch(void* const* d_in, const int* in_sizes, int n_in,
                              void* d_out, int out_size, void* d_ws, size_t ws_size,
                              hipStream_t stream) {
  const float* x  = (const float*)d_in[0];   // [32, 4096, 512]
  const float* w1 = (const float*)d_in[1];   // [512, 256]
  const float* w2 = (const float*)d_in[2];   // [256, 128]
  float* out = (float*)d_out;                // [32, 128]

  float* xm     = (float*)d_ws;                        // 4096*512 f32 (8 MB)
  float* cur    = xm  + (size_t)SEQ * EMBED;           // 4096*256 f32 (4 MB)
  float* counts = cur + (size_t)SEQ * HIDDEN;          // 256 f32

  k_batch_mean<<<(SEQ * EMBED) / 256, 256, 0, stream>>>(x, xm);

  dim3 g(SEQ / 16, HIDDEN / 16);       // 256 x 16 tiles, 1 wave each
  k_gemm_wmma<<<g, 32, 0, stream>>>(xm, w1, cur);

  k_scan<<<1, HIDDEN, 0, stream>>>(cur, counts);
  k_out<<<1, OUTD, 0, stream>>>(counts, w2, out);
}